// GroupedQuaryAttention_57758720197222
// MI455X (gfx1250) — compile-verified
//
#include <hip/hip_runtime.h>
#include <math.h>

// ---------------------------------------------------------------------------
// GroupedQueryAttention (bug-faithful) for MI455X / gfx1250, wave32 + WMMA.
//   B=2, S=2048, DIN=2048, HQ=16, HKV=1, DH=128.
// Reference collapses to:
//   K = q@Wk+bk, V = q@Wv+bv            [B*S,128]
//   S = (V K^T)/128, P = softmax_row(S), ctx = P V   (flash-style)
//   out = ctx @ Wo_sum + bo,  Wo_sum[d,n] = sum_h Wo[h*128+d, n]
// All matmuls use V_WMMA_F32_16X16X4_F32 (exact f32, matches f32 reference).
// GEMM kernels register-block 4 M-tiles/wave so each 2-load B-fragment feeds
// 4 WMMAs (loads per WMMA: 3 -> 1.5; VMEM issue rate was the ceiling since
// every B operand is L2-resident).
// ---------------------------------------------------------------------------

namespace {
constexpr int kB    = 2;
constexpr int kS    = 2048;
constexpr int kDin  = 2048;
constexpr int kDh   = 128;
constexpr int kHq   = 16;
constexpr int kDout = 2048;
constexpr int kMB   = 4;     // M-tiles (16 rows each) per wave in GEMM kernels
}

typedef __attribute__((ext_vector_type(2))) float v2f;
typedef __attribute__((ext_vector_type(8))) float v8f;

static __device__ __forceinline__ v8f wmma4(v2f a, v2f b, v8f c) {
  // D = A(16x4 f32) * B(4x16 f32) + C(16x16 f32)
  return __builtin_amdgcn_wmma_f32_16x16x4_f32(
      /*neg_a=*/false, a, /*neg_b=*/false, b,
      /*c_mod=*/(short)0, c, /*reuse_a=*/false, /*reuse_b=*/false);
}

// --------------------------- Kernel 0: Wo_sum ------------------------------
// Wo_sum[d, n] = sum_h Wo[h*128 + d, n]   (all 16 heads identical in ctx)
__global__ void gqa_wo_sum(const float* __restrict__ Wo,
                           float* __restrict__ WoSum) {
  int idx = blockIdx.x * blockDim.x + threadIdx.x;
  if (idx >= kDh * kDout) return;
  int d = idx / kDout;
  int n = idx % kDout;
  float s = 0.0f;
  #pragma unroll
  for (int h = 0; h < kHq; ++h)
    s += Wo[((size_t)(h * kDh + d)) * kDout + n];
  WoSum[idx] = s;
}

// --------------------------- Kernel 1: K/V proj ----------------------------
// One wave per 64x16 tile (4 stacked 16x16 WMMA tiles) of [B*S, 128].
__global__ void gqa_kv_proj(const float* __restrict__ q,
                            const float* __restrict__ Wk,
                            const float* __restrict__ bk,
                            const float* __restrict__ Wv,
                            const float* __restrict__ bv,
                            float* __restrict__ Kbuf,
                            float* __restrict__ Vbuf) {
  const int mt    = blockIdx.x;           // 0..63  (B*S / 64)
  const int nt    = blockIdx.y;           // 0..7   (128 / 16)
  const int which = blockIdx.z;           // 0 = K, 1 = V
  const float* __restrict__ W    = which ? Wv : Wk;
  const float* __restrict__ bias = which ? bv : bk;
  float* __restrict__ Out        = which ? Vbuf : Kbuf;

  const int lane = threadIdx.x;
  const int half = lane >> 4;             // lanes 16-31 hold K+2,K+3
  const int l16  = lane & 15;
  const int col  = nt * 16 + l16;         // B-fragment column (N)

  const float* __restrict__ arow[kMB];
  #pragma unroll
  for (int m = 0; m < kMB; ++m)
    arow[m] = q + (size_t)(mt * (16 * kMB) + m * 16 + l16) * kDin;

  v8f acc[kMB];
  #pragma unroll
  for (int m = 0; m < kMB; ++m) acc[m] = v8f{};

  #pragma unroll 4
  for (int k = 0; k < kDin; k += 4) {
    const int ka = k + half * 2;
    v2f b;                                                  // shared by 4 tiles
    b.x = W[(size_t)(ka    ) * kDh + col];                  // W[ka,   col]
    b.y = W[(size_t)(ka + 1) * kDh + col];                  // W[ka+1, col]
    #pragma unroll
    for (int m = 0; m < kMB; ++m) {
      v2f a = *(const v2f*)(arow[m] + ka);                  // q[row, ka..ka+1]
      acc[m] = wmma4(a, b, acc[m]);
    }
  }

  const float bb = bias[col];
  #pragma unroll
  for (int m = 0; m < kMB; ++m) {
    #pragma unroll
    for (int r = 0; r < 8; ++r) {
      const int mm = mt * (16 * kMB) + m * 16 + r + 8 * half; // C/D layout row
      Out[(size_t)mm * kDh + col] = acc[m][r] + bb;
    }
  }
}

// --------------------------- Kernel 2: attention ---------------------------
// Flash-style: one wave per 16-row block. A-frags (V rows) + 16x128 ctx in
// registers; P tile bounced through padded LDS for D->A layout conversion.
__global__ void gqa_attn(const float* __restrict__ Kbuf,
                         const float* __restrict__ Vbuf,
                         float* __restrict__ Ctx) {
  __shared__ float ldsP[16][17];          // padded: avoid 4-way bank conflict

  const int jt   = blockIdx.x;            // row block (0..127)
  const int b    = blockIdx.y;            // batch
  const int lane = threadIdx.x;
  const int half = lane >> 4;
  const int l16  = lane & 15;
  const int j0   = jt * 16;

  // A-fragments for the score GEMM: rows are V_j ("query" = V projection).
  const float* __restrict__ vrow = Vbuf + ((size_t)b * kS + j0 + l16) * kDh;
  v2f afrag[32];
  #pragma unroll
  for (int ks = 0; ks < 32; ++ks)
    afrag[ks] = *(const v2f*)(vrow + ks * 4 + half * 2);

  v8f ctx[8];
  #pragma unroll
  for (int n = 0; n < 8; ++n) ctx[n] = v8f{};

  // Online-softmax state; every lane tracks row (lane & 15) (halves duplicate).
  float mrow = -INFINITY;
  float lrow = 0.0f;

  const float* __restrict__ kbase = Kbuf + (size_t)b * kS * kDh;
  const float* __restrict__ vbase = Vbuf + (size_t)b * kS * kDh;

  for (int it = 0; it < kS / 16; ++it) {
    // prefetch next K tile row for this lane (global_prefetch_b8 path)
    if (it + 1 < kS / 16)
      __builtin_prefetch(kbase + (size_t)((it + 1) * 16 + l16) * kDh, 0, 1);

    // ---- scores tile: S = V_rows * K_rows^T / 128 ----
    const float* __restrict__ krow = kbase + (size_t)(it * 16 + l16) * kDh;
    v8f s = {};
    #pragma unroll
    for (int ks = 0; ks < 32; ++ks) {
      v2f bf = *(const v2f*)(krow + ks * 4 + half * 2);     // (K^T)[d, i]
      s = wmma4(afrag[ks], bf, s);
    }

    // D-layout -> LDS (row j = r + 8*half, col i = l16)
    #pragma unroll
    for (int r = 0; r < 8; ++r)
      ldsP[r + 8 * half][l16] = s[r] * (1.0f / 128.0f);
    __syncthreads();

    // ---- per-row online softmax (row = l16; both halves duplicate) ----
    float rowv[16];
    float mx = mrow;
    #pragma unroll
    for (int i = 0; i < 16; ++i) {
      rowv[i] = ldsP[l16][i];
      mx = fmaxf(mx, rowv[i]);
    }
    const float corr = __expf(mrow - mx);
    float psum = 0.0f;
    #pragma unroll
    for (int i = 0; i < 16; ++i) {
      rowv[i] = __expf(rowv[i] - mx);
      psum += rowv[i];
    }
    lrow = lrow * corr + psum;
    mrow = mx;
    __syncthreads();                      // all reads of S done before P write
    if (half == 0) {
      #pragma unroll
      for (int i = 0; i < 16; ++i) ldsP[l16][i] = rowv[i];
    }
    __syncthreads();

    // ---- rescale ctx accumulator: row j = r + 8*half has factor corr[j] ----
    #pragma unroll
    for (int r = 0; r < 8; ++r) {
      const float f = __shfl(corr, r + 8 * half);
      #pragma unroll
      for (int n = 0; n < 8; ++n) ctx[n][r] *= f;
    }

    // ---- ctx += P (16x16) @ V_block (16x128) ----
    #pragma unroll
    for (int n = 0; n < 8; ++n) {
      v8f c = ctx[n];
      #pragma unroll
      for (int ks = 0; ks < 4; ++ks) {
        const int kk = ks * 4 + half * 2;                   // local i index
        v2f a;                                              // A = P row l16
        a.x = ldsP[l16][kk];
        a.y = ldsP[l16][kk + 1];
        v2f bf;                                             // B = V[i, n]
        bf.x = vbase[(size_t)(it * 16 + kk    ) * kDh + n * 16 + l16];
        bf.y = vbase[(size_t)(it * 16 + kk + 1) * kDh + n * 16 + l16];
        c = wmma4(a, bf, c);
      }
      ctx[n] = c;
    }
    __syncthreads();                      // P tile consumed before next store
  }

  // ---- finalize: divide by row sums, store ctx ----
  #pragma unroll
  for (int r = 0; r < 8; ++r) {
    const float linv = 1.0f / __shfl(lrow, r + 8 * half);
    const int j = j0 + r + 8 * half;
    #pragma unroll
    for (int n = 0; n < 8; ++n)
      Ctx[((size_t)b * kS + j) * kDh + n * 16 + l16] = ctx[n][r] * linv;
  }
}

// --------------------------- Kernel 3: out proj ----------------------------
// out[m, n] = ctx[m, :] @ Wo_sum[:, n] + bo[n].  One wave per 64x16 tile.
__global__ void gqa_out_proj(const float* __restrict__ Ctx,
                             const float* __restrict__ WoSum,
                             const float* __restrict__ bo,
                             float* __restrict__ out) {
  const int mt   = blockIdx.x;            // 0..63  (B*S / 64)
  const int nt   = blockIdx.y;            // 0..127 (2048 / 16)
  const int lane = threadIdx.x;
  const int half = lane >> 4;
  const int l16  = lane & 15;
  const int col  = nt * 16 + l16;

  const float* __restrict__ arow[kMB];
  #pragma unroll
  for (int m = 0; m < kMB; ++m)
    arow[m] = Ctx + (size_t)(mt * (16 * kMB) + m * 16 + l16) * kDh;

  v8f acc[kMB];
  #pragma unroll
  for (int m = 0; m < kMB; ++m) acc[m] = v8f{};

  #pragma unroll
  for (int ks = 0; ks < 32; ++ks) {
    const int k = ks * 4 + half * 2;
    v2f b;                                                  // shared by 4 tiles
    b.x = WoSum[(size_t)(k    ) * kDout + col];
    b.y = WoSum[(size_t)(k + 1) * kDout + col];
    #pragma unroll
    for (int m = 0; m < kMB; ++m) {
      v2f a = *(const v2f*)(arow[m] + k);
      acc[m] = wmma4(a, b, acc[m]);
    }
  }

  const float bb = bo[col];
  #pragma unroll
  for (int m = 0; m < kMB; ++m) {
    #pragma unroll
    for (int r = 0; r < 8; ++r) {
      const int mm = mt * (16 * kMB) + m * 16 + r + 8 * half;
      out[(size_t)mm * kDout + col] = acc[m][r] + bb;
    }
  }
}

// ------------------------------- launcher ----------------------------------
extern "C" void kernel_launch(void* const* d_in, const int* in_sizes, int n_in,
                              void* d_out, int out_size, void* d_ws, size_t ws_size,
                              hipStream_t stream) {
  (void)in_sizes; (void)n_in; (void)out_size; (void)ws_size;
  const float* q  = (const float*)d_in[0];
  // d_in[1] = Wq, d_in[2] = bq : dead in the bug-faithful reference.
  const float* Wk = (const float*)d_in[3];
  const float* bk = (const float*)d_in[4];
  const float* Wv = (const float*)d_in[5];
  const float* bv = (const float*)d_in[6];
  const float* Wo = (const float*)d_in[7];
  const float* bo = (const float*)d_in[8];
  float* out = (float*)d_out;

  // Workspace layout (floats): Kbuf | Vbuf | Ctx | Wo_sum  (7 MB total)
  float* ws    = (float*)d_ws;
  float* Kbuf  = ws;                                   // B*S*DH = 524288
  float* Vbuf  = ws + (size_t)kB * kS * kDh;           // 524288
  float* Ctx   = ws + (size_t)2 * kB * kS * kDh;       // 524288
  float* WoSum = ws + (size_t)3 * kB * kS * kDh;       // 128*2048 = 262144

  // 0) collapse Wo over the 16 identical heads
  {
    const int n = kDh * kDout;
    gqa_wo_sum<<<(n + 255) / 256, 256, 0, stream>>>(Wo, WoSum);
  }
  // 1) K/V projections (WMMA f32): grid = (M/64, 128/16, {K,V})
  gqa_kv_proj<<<dim3(kB * kS / (16 * kMB), kDh / 16, 2), 32, 0, stream>>>(
      q, Wk, bk, Wv, bv, Kbuf, Vbuf);
  // 2) flash attention: one wave per 16-row block per batch
  gqa_attn<<<dim3(kS / 16, kB), 32, 0, stream>>>(Kbuf, Vbuf, Ctx);
  // 3) output projection + bias
  gqa_out_proj<<<dim3(kB * kS / (16 * kMB), kDout / 16), 32, 0, stream>>>(
      Ctx, WoSum, bo, out);
}